// CompGcnBasis_86045374808383
// MI455X (gfx1250) — compile-verified
//
#include <hip/hip_runtime.h>

typedef __attribute__((ext_vector_type(2))) float v2f;
typedef __attribute__((ext_vector_type(8))) float v8f;

#define NN       100000   // nodes
#define IC       200      // in channels
#define OC       200      // out channels
#define NB       50       // basis
#define REL_ROWS 475      // 2*237 relations + loop row
#define RELP     480      // rel_emb rows padded to 16
#define NE       400000   // directed edges per half
#define NPAD     208      // OC padded to 13*16
#define NT13     13       // N tiles

// ---------------- zero fill ----------------
__global__ void zero4_kernel(float4* p, int n4) {
    int i = blockIdx.x * blockDim.x + threadIdx.x;
    if (i < n4) p[i] = make_float4(0.f, 0.f, 0.f, 0.f);
}

// ---------------- pad B (IC x OC) -> (IC x NPAD), zero-filled ----------------
__global__ void padB_kernel(const float* __restrict__ src, float* __restrict__ dst) {
    int i = blockIdx.x * blockDim.x + threadIdx.x;
    if (i >= IC * NPAD) return;
    int k = i / NPAD, c = i % NPAD;
    dst[i] = (c < OC) ? src[k * OC + c] : 0.f;
}

// ---------------- rel_emb = concat(rel_weight @ basis_vector, loop_rel), padded rows ----------------
__global__ void relemb_kernel(const float* __restrict__ rw, const float* __restrict__ bv,
                              const float* __restrict__ loop, float* __restrict__ rel) {
    int i = blockIdx.x * blockDim.x + threadIdx.x;
    if (i >= RELP * IC) return;
    int r = i / IC, c = i % IC;
    if (r < REL_ROWS - 1) {
        float s = 0.f;
        #pragma unroll 10
        for (int b = 0; b < NB; ++b) s += rw[r * NB + b] * bv[b * IC + c];
        rel[i] = s;
    } else if (r == REL_ROWS - 1) {
        rel[i] = loop[c];
    } else {
        rel[i] = 0.f;                 // pad rows 475..479
    }
}

// ---------------- bias[n] = -(loop_rel @ w_loop)[n] ----------------
__global__ void bias_kernel(const float* __restrict__ loop, const float* __restrict__ wl,
                            float* __restrict__ bias) {
    int n = blockIdx.x * blockDim.x + threadIdx.x;
    if (n >= OC) return;
    float s = 0.f;
    for (int k = 0; k < IC; ++k) s += loop[k] * wl[k * OC + n];
    bias[n] = -s;
}

// ---------------- head degrees (float, per direction) ----------------
__global__ void degree_kernel(const int* __restrict__ heads,
                              float* __restrict__ deg_in, float* __restrict__ deg_out) {
    int e = blockIdx.x * blockDim.x + threadIdx.x;
    if (e >= 2 * NE) return;
    int h = heads[e];
    float* deg = (e < NE) ? deg_in : deg_out;
    __hip_atomic_fetch_add(&deg[h], 1.0f, __ATOMIC_RELAXED, __HIP_MEMORY_SCOPE_AGENT);
}

// ---------------- deg -> rsqrt(deg) in place ----------------
__global__ void dinv_kernel(float* __restrict__ deg, int n) {
    int i = blockIdx.x * blockDim.x + threadIdx.x;
    if (i >= n) return;
    float d = deg[i];
    deg[i] = (d > 0.f) ? __frsqrt_rn(d) : 0.f;
}

// ---------------- per-edge aggregation: agg[h] += norm*(x[t] - rel_emb[r]) ----------------
__global__ void aggregate_kernel(const int* __restrict__ heads, const int* __restrict__ tails,
                                 const int* __restrict__ etype,
                                 const float* __restrict__ x, const float* __restrict__ rel,
                                 const float* __restrict__ dinv_in, const float* __restrict__ dinv_out,
                                 float* __restrict__ agg_in, float* __restrict__ agg_out) {
    int wave = blockIdx.x * (blockDim.x >> 5) + (threadIdx.x >> 5);
    int lane = threadIdx.x & 31;
    if (wave >= 2 * NE) return;
    int h = heads[wave];
    int t = tails[wave];
    int r = etype[wave];
    const float* dinv = (wave < NE) ? dinv_in : dinv_out;
    float* agg        = (wave < NE) ? agg_in  : agg_out;
    float norm = dinv[h] * dinv[t];
    const float* xt = x   + (long)t * IC;
    const float* re = rel + (long)r * IC;
    float*       ah = agg + (long)h * IC;
    for (int c = lane; c < IC; c += 32) {
        float v = norm * (xt[c] - re[c]);
        __hip_atomic_fetch_add(&ah[c], v, __ATOMIC_RELAXED, __HIP_MEMORY_SCOPE_AGENT);
    }
}

// ---------------- fused D = tanh((A0@B0 + A1@B1 + A2@B2 + bias) / 3) ----------------
// one wave = 16 rows x 4 N-tiles (A fragments reused 4x); B padded to NPAD cols
__global__ void gemm3_tanh_kernel(const float* __restrict__ A0, const float* __restrict__ A1,
                                  const float* __restrict__ A2,
                                  const float* __restrict__ B0, const float* __restrict__ B1,
                                  const float* __restrict__ B2,
                                  const float* __restrict__ bias, float* __restrict__ out,
                                  int Mtiles) {
    int wave  = blockIdx.x * (blockDim.x >> 5) + (threadIdx.x >> 5);
    int lane  = threadIdx.x & 31;
    int tm    = wave >> 2;                    // 4 chunks of N-tiles per M-row
    int chunk = wave & 3;
    if (tm >= Mtiles) return;                 // whole-wave exit: EXEC all-ones for WMMA

    int l15   = lane & 15;
    int khalf = (lane >> 4) << 1;             // 0 for lanes 0-15, 2 for lanes 16-31
    int arow  = tm * 16 + l15;
    int tn0   = chunk * 4;

    const float* a0p = A0 + (long)arow * IC + khalf;
    const float* a1p = A1 + (long)arow * IC + khalf;
    const float* a2p = A2 + (long)arow * IC + khalf;

    int col[4];
    #pragma unroll
    for (int j = 0; j < 4; ++j) {
        int tnj = tn0 + j; if (tnj > NT13 - 1) tnj = NT13 - 1;   // clamp tail (dup compute, store masked)
        col[j] = tnj * 16 + l15;
    }

    v8f acc[4] = {};
    for (int k = 0; k < IC; k += 4) {
        v2f a0 = *(const v2f*)(a0p + k);
        v2f a1 = *(const v2f*)(a1p + k);
        v2f a2 = *(const v2f*)(a2p + k);
        int kk = k + khalf;
        const float* b0r = B0 + (long)kk * NPAD;
        const float* b1r = B1 + (long)kk * NPAD;
        const float* b2r = B2 + (long)kk * NPAD;
        #pragma unroll
        for (int j = 0; j < 4; ++j) {
            v2f b0 = { b0r[col[j]], b0r[NPAD + col[j]] };
            v2f b1 = { b1r[col[j]], b1r[NPAD + col[j]] };
            v2f b2 = { b2r[col[j]], b2r[NPAD + col[j]] };
            acc[j] = __builtin_amdgcn_wmma_f32_16x16x4_f32(false, a0, false, b0, (short)0, acc[j], false, false);
            acc[j] = __builtin_amdgcn_wmma_f32_16x16x4_f32(false, a1, false, b1, (short)0, acc[j], false, false);
            acc[j] = __builtin_amdgcn_wmma_f32_16x16x4_f32(false, a2, false, b2, (short)0, acc[j], false, false);
        }
    }

    int mbase = tm * 16 + ((lane >> 4) << 3);   // lanes 16-31 hold rows M=r+8
    #pragma unroll
    for (int j = 0; j < 4; ++j) {
        int tnj = tn0 + j;
        if (tnj < NT13) {
            int c = tnj * 16 + l15;
            if (c < OC) {
                float bn = bias[c];
                #pragma unroll
                for (int r = 0; r < 8; ++r) {
                    float v = tanhf((acc[j][r] + bn) * (1.0f / 3.0f));
                    __builtin_nontemporal_store(v, &out[(long)(mbase + r) * OC + c]);
                }
            }
        }
    }
}

// ---------------- D = A @ B for out_2; A row-padded (RELP), B col-padded (NPAD) ----------------
__global__ void gemm_wmma_kernel(const float* __restrict__ A, const float* __restrict__ B,
                                 float* __restrict__ D, int M, int Mtiles) {
    int wave = blockIdx.x * (blockDim.x >> 5) + (threadIdx.x >> 5);
    int lane = threadIdx.x & 31;
    int tm = wave / NT13;
    int tn = wave % NT13;
    if (tm >= Mtiles) return;

    int l15   = lane & 15;
    int khalf = (lane >> 4) << 1;
    int arow  = tm * 16 + l15;                // < RELP, always valid (padded rows are zero)
    int col   = tn * 16 + l15;                // < NPAD, always valid (padded cols are zero)

    const float* ap = A + (long)arow * IC + khalf;
    v8f acc = {};
    for (int k = 0; k < IC; k += 4) {
        v2f a = *(const v2f*)(ap + k);
        int kk = k + khalf;
        v2f b = { B[(long)kk * NPAD + col], B[(long)(kk + 1) * NPAD + col] };
        acc = __builtin_amdgcn_wmma_f32_16x16x4_f32(false, a, false, b, (short)0, acc, false, false);
    }

    if (col < OC) {
        int mbase = tm * 16 + ((lane >> 4) << 3);
        #pragma unroll
        for (int r = 0; r < 8; ++r) {
            int m = mbase + r;
            if (m < M) D[(long)m * OC + col] = acc[r];
        }
    }
}

extern "C" void kernel_launch(void* const* d_in, const int* in_sizes, int n_in,
                              void* d_out, int out_size, void* d_ws, size_t ws_size,
                              hipStream_t stream) {
    const float* x     = (const float*)d_in[0];
    const int*   eidx  = (const int*)  d_in[1];   // (2, 800000): row0 = heads, row1 = tails
    const int*   etyp  = (const int*)  d_in[2];
    const float* bv    = (const float*)d_in[3];   // (50, 200)
    const float* rw    = (const float*)d_in[4];   // (474, 50)
    const float* wrel  = (const float*)d_in[5];   // (200, 200)
    const float* loop  = (const float*)d_in[6];   // (1, 200)
    const float* win   = (const float*)d_in[7];
    const float* wout  = (const float*)d_in[8];
    const float* wloop = (const float*)d_in[9];

    float* out  = (float*)d_out;                  // (100000, 200)
    float* out2 = out + (size_t)NN * OC;          // (475, 200)

    // workspace layout (floats)
    float* ws      = (float*)d_ws;
    float* deg_in  = ws;                                  // NN
    float* deg_out = ws + NN;                             // NN
    float* agg_in  = ws + 2 * (size_t)NN;                 // NN*IC
    float* agg_out = agg_in + (size_t)NN * IC;            // NN*IC
    float* rel_emb = agg_out + (size_t)NN * IC;           // RELP*IC
    float* biasv   = rel_emb + (size_t)RELP * IC;         // 256 (padded slot)
    float* winP    = biasv + 256;                         // IC*NPAD each
    float* woutP   = winP  + (size_t)IC * NPAD;
    float* wloopP  = woutP + (size_t)IC * NPAD;
    float* wrelP   = wloopP + (size_t)IC * NPAD;

    const int* heads = eidx;
    const int* tails = eidx + 2 * NE;

    // 1. zero deg + agg (40,200,000 floats, divisible by 4)
    int zn4 = (2 * NN + 2 * NN * IC) / 4;
    zero4_kernel<<<(zn4 + 255) / 256, 256, 0, stream>>>((float4*)ws, zn4);

    // 2. rel_emb (row-padded), bias, padded B matrices
    relemb_kernel<<<(RELP * IC + 255) / 256, 256, 0, stream>>>(rw, bv, loop, rel_emb);
    bias_kernel<<<1, 256, 0, stream>>>(loop, wloop, biasv);
    int pb = (IC * NPAD + 255) / 256;
    padB_kernel<<<pb, 256, 0, stream>>>(win,   winP);
    padB_kernel<<<pb, 256, 0, stream>>>(wout,  woutP);
    padB_kernel<<<pb, 256, 0, stream>>>(wloop, wloopP);
    padB_kernel<<<pb, 256, 0, stream>>>(wrel,  wrelP);

    // 3. degrees, 4. rsqrt in place
    degree_kernel<<<(2 * NE + 255) / 256, 256, 0, stream>>>(heads, deg_in, deg_out);
    dinv_kernel<<<(2 * NN + 255) / 256, 256, 0, stream>>>(ws, 2 * NN);

    // 5. edge aggregation: one wave per edge
    {
        long threads = (long)2 * NE * 32;
        aggregate_kernel<<<(int)((threads + 255) / 256), 256, 0, stream>>>(
            heads, tails, etyp, x, rel_emb, deg_in, deg_out, agg_in, agg_out);
    }

    // 6. fused WMMA GEMM x3 + bias + tanh (4 N-tiles per wave)
    int Mt = NN / 16;            // 6250
    {
        long waves = (long)Mt * 4;
        gemm3_tanh_kernel<<<(int)((waves + 7) / 8), 256, 0, stream>>>(
            agg_in, agg_out, x, winP, woutP, wloopP, biasv, out, Mt);
    }

    // 7. out_2 = rel_emb @ weight_rel
    int Mt2 = RELP / 16;         // 30
    {
        long waves = (long)Mt2 * NT13;
        gemm_wmma_kernel<<<(int)((waves + 7) / 8), 256, 0, stream>>>(
            rel_emb, wrelP, out2, REL_ROWS, Mt2);
    }
}